// MLP_84980222919390
// MI455X (gfx1250) — compile-verified
//
#include <hip/hip_runtime.h>
#include <math.h>

typedef float v2f __attribute__((ext_vector_type(2)));
typedef float v4f __attribute__((ext_vector_type(4)));
typedef float v8f __attribute__((ext_vector_type(8)));

#define B_  128
#define IN_ 512
#define H_  1024

__device__ __forceinline__ float fast_tanh(float x) {
#if __has_builtin(__builtin_amdgcn_tanhf)
    return __builtin_amdgcn_tanhf(x);          // gfx1250 v_tanh_f32
#elif __has_builtin(__builtin_amdgcn_tanh_f32)
    return __builtin_amdgcn_tanh_f32(x);
#else
    return tanhf(x);
#endif
}

// ---------------------------------------------------------------------------
// WMMA f32 GEMM: D[M,N] = act(A[M,K] @ W[N,K]^T + bias[N])
// One wave (32 lanes) per 16x16 output tile, K-loop step 4 via
// V_WMMA_F32_16X16X4_F32.
//   A fragment (16x4): lane&15 = M row, lane>>4 selects K-pair {0,1}/{2,3}
//   B fragment (4x16): lane&15 = N col (B[k,n] = W[n,k] -> same row-major load)
//   C/D: VGPR i -> row m0 + i + 8*(lane>>4), col n0 + (lane&15)
// ---------------------------------------------------------------------------
template <int ACT>
__global__ __launch_bounds__(32) void wmma_gemm(const float* __restrict__ A,
                                                const float* __restrict__ W,
                                                const float* __restrict__ bias,
                                                float* __restrict__ D,
                                                int N, int K) {
    const int tiles_n = N >> 4;
    const int tile = blockIdx.x;
    const int m0 = (tile / tiles_n) << 4;
    const int n0 = (tile % tiles_n) << 4;
    const int lane = threadIdx.x;
    const int half = lane >> 4;
    const int r = lane & 15;

    const float* Ar = A + (size_t)(m0 + r) * K + 2 * half;
    const float* Wr = W + (size_t)(n0 + r) * K + 2 * half;

    v8f acc = {};
    for (int k = 0; k < K; k += 4) {
        v2f a = *(const v2f*)(Ar + k);
        v2f b = *(const v2f*)(Wr + k);
        acc = __builtin_amdgcn_wmma_f32_16x16x4_f32(
            false, a, false, b, (short)0, acc, false, false);
    }

    const float bb = bias[n0 + r];
#pragma unroll
    for (int i = 0; i < 8; ++i) {
        float v = acc[i] + bb;
        if (ACT) v = fast_tanh(v);
        D[(size_t)(m0 + i + 8 * half) * N + n0 + r] = v;
    }
}

// t3 = tanh((hidden + reward*Wr + br) @ W3^T + b3), reward-embedding fused
// into the A-fragment load.
__global__ __launch_bounds__(32) void wmma_gemm_h3(const float* __restrict__ hidden,
                                                   const float* __restrict__ reward,
                                                   const float* __restrict__ WrV,
                                                   const float* __restrict__ brV,
                                                   const float* __restrict__ W3,
                                                   const float* __restrict__ b3,
                                                   float* __restrict__ t3) {
    const int N = H_, K = H_;
    const int tiles_n = N >> 4;
    const int tile = blockIdx.x;
    const int m0 = (tile / tiles_n) << 4;
    const int n0 = (tile % tiles_n) << 4;
    const int lane = threadIdx.x;
    const int half = lane >> 4;
    const int r = lane & 15;

    const float rw = reward[m0 + r];
    const float* Ar = hidden + (size_t)(m0 + r) * K + 2 * half;
    const float* Wrow = W3 + (size_t)(n0 + r) * K + 2 * half;
    const float* wrp = WrV + 2 * half;
    const float* brp = brV + 2 * half;

    v8f acc = {};
    for (int k = 0; k < K; k += 4) {
        v2f hh = *(const v2f*)(Ar + k);
        v2f w2 = *(const v2f*)(wrp + k);
        v2f br2 = *(const v2f*)(brp + k);
        v2f a;
        a.x = hh.x + rw * w2.x + br2.x;
        a.y = hh.y + rw * w2.y + br2.y;
        v2f b = *(const v2f*)(Wrow + k);
        acc = __builtin_amdgcn_wmma_f32_16x16x4_f32(
            false, a, false, b, (short)0, acc, false, false);
    }

    const float bb = b3[n0 + r];
#pragma unroll
    for (int i = 0; i < 8; ++i) {
        t3[(size_t)(m0 + i + 8 * half) * N + n0 + r] = fast_tanh(acc[i] + bb);
    }
}

// ---------------------------------------------------------------------------
// Pass 1 over plastic (512 MB, NT loads): contrib[b,h] = sum_i a[h,i]p[b,h,i]e[b,i]
// pre = current + contrib; hidden = tanh(pre). One wave32 per (b,h) row.
// ---------------------------------------------------------------------------
__global__ __launch_bounds__(256) void contrib_kernel(const float* __restrict__ plastic,
                                                      const float* __restrict__ alpha,
                                                      const float* __restrict__ emb,
                                                      const float* __restrict__ cur,
                                                      float* __restrict__ pre_out,
                                                      float* __restrict__ hidden_out) {
    const int row = blockIdx.x * 8 + (threadIdx.x >> 5);   // b*H + h
    const int lane = threadIdx.x & 31;
    const int b = row >> 10;
    const int h = row & (H_ - 1);

    const v4f* pl = (const v4f*)(plastic + ((size_t)row << 10));
    const v4f* al = (const v4f*)(alpha + ((size_t)h << 10));
    const v4f* em = (const v4f*)(emb + ((size_t)b << 10));

    float s = 0.0f;
    for (int j = lane; j < H_ / 4; j += 32) {
        v4f p = __builtin_nontemporal_load(pl + j);   // streaming, bypass cache
        v4f a = al[j];
        v4f e = em[j];
        s += p.x * a.x * e.x + p.y * a.y * e.y + p.z * a.z * e.z + p.w * a.w * e.w;
    }
#pragma unroll
    for (int off = 16; off > 0; off >>= 1) s += __shfl_down(s, off, 32);

    if (lane == 0) {
        const float pre = cur[row] + s;
        pre_out[row] = pre;
        hidden_out[row] = fast_tanh(pre);
    }
}

// choice/value/neuromodulator heads: one wave32 per batch row.
__global__ __launch_bounds__(256) void heads_kernel(const float* __restrict__ hidden,
                                                    const float* __restrict__ t3,
                                                    const float* __restrict__ Wc,
                                                    const float* __restrict__ bc,
                                                    const float* __restrict__ Wv,
                                                    const float* __restrict__ bv,
                                                    const float* __restrict__ Wn,
                                                    const float* __restrict__ bn,
                                                    float* __restrict__ o_choice,
                                                    float* __restrict__ o_nm,
                                                    float* __restrict__ o_value) {
    const int b = blockIdx.x * 8 + (threadIdx.x >> 5);
    const int lane = threadIdx.x & 31;

    const v4f* hv = (const v4f*)(hidden + ((size_t)b << 10));
    const v4f* tv = (const v4f*)(t3 + ((size_t)b << 10));
    const v4f* wc = (const v4f*)Wc;
    const v4f* wv = (const v4f*)Wv;
    const v4f* wn0 = (const v4f*)Wn;
    const v4f* wn1 = (const v4f*)(Wn + H_);

    float sc = 0.f, sv = 0.f, s0 = 0.f, s1 = 0.f;
    for (int j = lane; j < H_ / 4; j += 32) {
        v4f h = hv[j], t = tv[j];
        v4f c = wc[j], v = wv[j], n0 = wn0[j], n1 = wn1[j];
        sc += h.x * c.x + h.y * c.y + h.z * c.z + h.w * c.w;
        sv += h.x * v.x + h.y * v.y + h.z * v.z + h.w * v.w;
        s0 += t.x * n0.x + t.y * n0.y + t.z * n0.z + t.w * n0.w;
        s1 += t.x * n1.x + t.y * n1.y + t.z * n1.z + t.w * n1.w;
    }
#pragma unroll
    for (int off = 16; off > 0; off >>= 1) {
        sc += __shfl_down(sc, off, 32);
        sv += __shfl_down(sv, off, 32);
        s0 += __shfl_down(s0, off, 32);
        s1 += __shfl_down(s1, off, 32);
    }
    if (lane == 0) {
        o_choice[b] = 1.0f / (1.0f + __expf(-(sc + bc[0])));
        o_value[b] = sv + bv[0];
        o_nm[b] = fast_tanh(s0 + bn[0]) - fast_tanh(s1 + bn[1]);
    }
}

// ---------------------------------------------------------------------------
// Pass 2 over plastic (NT load + NT store, 1 GB traffic):
// new_p = clip(p + nm[b] * 10*tanh(pre[b,h]*emb[b,i]), -50, 50)
// ---------------------------------------------------------------------------
__global__ __launch_bounds__(256) void plastic_update_kernel(const float* __restrict__ plastic,
                                                             const float* __restrict__ pre,
                                                             const float* __restrict__ emb,
                                                             const float* __restrict__ nm,
                                                             float* __restrict__ outp) {
    const size_t tid = (size_t)blockIdx.x * 256 + threadIdx.x;  // one float4 each
    const int row = (int)(tid >> 8);          // b*H + h
    const int i = (int)(tid & 255) << 2;      // i
    const int b = row >> 10;

    const float pr = pre[row];
    const float g = nm[b] * 10.0f;
    const v4f e = *(const v4f*)(emb + ((size_t)b << 10) + i);
    const size_t off = ((size_t)row << 10) + i;
    v4f p = __builtin_nontemporal_load((const v4f*)(plastic + off));

    v4f r;
    r.x = fminf(fmaxf(p.x + g * fast_tanh(pr * e.x), -50.0f), 50.0f);
    r.y = fminf(fmaxf(p.y + g * fast_tanh(pr * e.y), -50.0f), 50.0f);
    r.z = fminf(fmaxf(p.z + g * fast_tanh(pr * e.z), -50.0f), 50.0f);
    r.w = fminf(fmaxf(p.w + g * fast_tanh(pr * e.w), -50.0f), 50.0f);
    __builtin_nontemporal_store(r, (v4f*)(outp + off));
}

extern "C" void kernel_launch(void* const* d_in, const int* in_sizes, int n_in,
                              void* d_out, int out_size, void* d_ws, size_t ws_size,
                              hipStream_t stream) {
    (void)in_sizes; (void)n_in; (void)out_size; (void)ws_size;

    const float* items   = (const float*)d_in[0];
    const float* plastic = (const float*)d_in[1];
    const float* reward  = (const float*)d_in[2];
    const float* W1 = (const float*)d_in[3];
    const float* b1 = (const float*)d_in[4];
    const float* W2 = (const float*)d_in[5];
    const float* b2 = (const float*)d_in[6];
    const float* W3 = (const float*)d_in[7];
    const float* b3 = (const float*)d_in[8];
    const float* Wc = (const float*)d_in[9];
    const float* bc = (const float*)d_in[10];
    const float* Wr = (const float*)d_in[11];
    const float* br = (const float*)d_in[12];
    const float* Wn = (const float*)d_in[13];
    const float* bn = (const float*)d_in[14];
    const float* alpha = (const float*)d_in[15];
    const float* Wv = (const float*)d_in[16];
    const float* bv = (const float*)d_in[17];

    float* out = (float*)d_out;
    float* o_choice = out;                                  // [B]
    float* o_nm     = out + B_;                             // [B]
    float* o_value  = out + 2 * B_;                         // [B]
    float* o_plast  = out + 3 * B_;                         // [B,H,H]
    float* o_hidden = o_plast + (size_t)B_ * H_ * H_;       // [B,H]

    float* ws = (float*)d_ws;
    float* w_emb = ws;                  // [B,H]
    float* w_cur = w_emb + B_ * H_;     // [B,H]
    float* w_pre = w_cur + B_ * H_;     // [B,H]
    float* w_t3  = w_pre + B_ * H_;     // [B,H]

    const int gemm_tiles = (B_ / 16) * (H_ / 16);           // 512 waves

    // 1) emb = tanh(items @ W1^T + b1)
    wmma_gemm<1><<<gemm_tiles, 32, 0, stream>>>(items, W1, b1, w_emb, H_, IN_);
    // 2) current = emb @ W2^T + b2
    wmma_gemm<0><<<gemm_tiles, 32, 0, stream>>>(w_emb, W2, b2, w_cur, H_, H_);
    // 3) plastic contrib -> pre_tanh_hidden, hidden   (pass 1 over 512 MB)
    contrib_kernel<<<(B_ * H_) / 8, 256, 0, stream>>>(plastic, alpha, w_emb, w_cur,
                                                      w_pre, o_hidden);
    // 4) t3 = tanh((hidden + reward_emb) @ W3^T + b3)
    wmma_gemm_h3<<<gemm_tiles, 32, 0, stream>>>(o_hidden, reward, Wr, br, W3, b3, w_t3);
    // 5) choice / neuromodulator / value heads
    heads_kernel<<<B_ / 8, 256, 0, stream>>>(o_hidden, w_t3, Wc, bc, Wv, bv, Wn, bn,
                                             o_choice, o_nm, o_value);
    // 6) new_plastic (pass 2: 512 MB read + 512 MB write, NT)
    const int upd_blocks = (int)(((size_t)B_ * H_ * H_ / 4) / 256);  // 131072
    plastic_update_kernel<<<upd_blocks, 256, 0, stream>>>(plastic, w_pre, w_emb,
                                                          o_nm, o_plast);
}